// FK2D_81604378624334
// MI455X (gfx1250) — compile-verified
//
#include <hip/hip_runtime.h>

typedef __attribute__((ext_vector_type(2))) float v2f;
typedef __attribute__((ext_vector_type(8))) float v8f;

#define T_TOTAL 262144
#define NJ 49
#define NWAVES 2                 // 64-thread blocks: 2 independent waves
#define TBLOCK (NWAVES * 32)     // 64 timesteps per block
#define KSTRIDE 72               // words per joint row in LDS: 64 used + 8 pad
                                 // (2*72 mod 64 == 16 -> lane halves hit disjoint banks)

// FK2D: one timestep per lane computes deltas (serial cumsum of angles + sincos),
// then the 49-joint prefix sum runs as L(tri-ones) x D on the WMMA pipe in f32.
__global__ __launch_bounds__(TBLOCK) void fk2d_kernel(
    const float* __restrict__ thetas,      // (T, 49)
    const float* __restrict__ rootpos,     // (T, 49, 2)  only joint 0 used
    const float* __restrict__ scales,      // (T, 49)
    const float* __restrict__ rest,        // (49, 2)
    float* __restrict__ out)               // (T, 49, 2)
{
    __shared__ float ldsD[NWAVES][NJ][KSTRIDE];

    const int lane  = threadIdx.x & 31;
    const int wave  = threadIdx.x >> 5;
    const int tBase = blockIdx.x * TBLOCK + wave * 32;
    const int t     = tBase + lane;

    float (*D)[KSTRIDE] = ldsD[wave];

    // ---------- Phase 1: per-lane delta vectors into LDS ----------
    // Smoothed root position is the k=0 "delta" so L x D includes it for free.
    float px = rootpos[t * 98 + 0];
    float py = rootpos[t * 98 + 1];
    if (t > 0 && t < T_TOTAL - 1) {
        float pxm = rootpos[(t - 1) * 98 + 0];
        float pym = rootpos[(t - 1) * 98 + 1];
        float pxp = rootpos[(t + 1) * 98 + 0];
        float pyp = rootpos[(t + 1) * 98 + 1];
        px = (pxm + 4.0f * px + pxp) * (1.0f / 6.0f);
        py = (pym + 4.0f * py + pyp) * (1.0f / 6.0f);
    }
    D[0][lane * 2 + 0] = px;
    D[0][lane * 2 + 1] = py;

    const float* thRow = thetas + t * NJ;
    const float* bsRow = scales + t * NJ;
    float phi = 0.0f;
    #pragma unroll 4
    for (int j = 1; j < NJ; ++j) {
        phi += thRow[j];
        float s, c;
        __sincosf(phi, &s, &c);
        float bs = bsRow[j];
        float ox = rest[2 * j + 0] * bs;
        float oy = rest[2 * j + 1] * bs;
        D[j][lane * 2 + 0] = c * ox - s * oy;   // R(phi) * off
        D[j][lane * 2 + 1] = s * ox + c * oy;
    }

    // Cross-lane LDS handoff within this wave only: DScnt drain is sufficient.
    __builtin_amdgcn_wave_barrier();
    asm volatile("s_wait_dscnt 0" ::: "memory");
    __builtin_amdgcn_wave_barrier();

    // ---------- Phase 2: prefix sum over joints via f32 WMMA ----------
    // Columns n = tt*2 + comp (8 timesteps x 2 components per 16-wide tile).
    const int n    = lane & 15;
    const int half = lane >> 4;      // A/B K-split: lanes 16-31 hold K+2
    const int tt   = n >> 1;
    const int comp = n & 1;

    for (int b = 0; b < 4; ++b) {    // 4 batches of 8 timesteps per wave
        const int tl = b * 8 + tt;   // local timestep 0..31

        v2f Bv[13];                  // D-matrix tiles for all K-steps (K = 52, 49 valid)
        #pragma unroll
        for (int kk = 0; kk < 13; ++kk) {
            int k0  = kk * 4 + half * 2;
            int k1  = k0 + 1;
            int k0c = (k0 < NJ) ? k0 : 0;          // clamp addr, mask value:
            int k1c = (k1 < NJ) ? k1 : 0;          // keeps EXEC uniform (cndmask)
            float b0 = D[k0c][tl * 2 + comp];
            float b1 = D[k1c][tl * 2 + comp];
            Bv[kk][0] = (k0 < NJ) ? b0 : 0.0f;
            Bv[kk][1] = (k1 < NJ) ? b1 : 0.0f;
        }

        #pragma unroll
        for (int mt = 0; mt < 4; ++mt) {           // 4 M-tiles cover joints 0..63
            v8f acc = {};
            const int kkMax = (mt == 3) ? 13 : (mt + 1) * 4;  // skip all-zero tri tiles
            #pragma unroll
            for (int kk = 0; kk < kkMax; ++kk) {
                int m  = mt * 16 + (lane & 15);
                int k0 = kk * 4 + half * 2;
                v2f Av;                             // lower-triangular ones, on the fly
                Av[0] = (m >= k0)     ? 1.0f : 0.0f;
                Av[1] = (m >= k0 + 1) ? 1.0f : 0.0f;
                acc = __builtin_amdgcn_wmma_f32_16x16x4_f32(
                    false, Av, false, Bv[kk], (short)0, acc, false, false);
            }
            const int tg = tBase + tl;
            #pragma unroll
            for (int r = 0; r < 8; ++r) {
                int m = mt * 16 + r + half * 8;     // C/D layout: VGPR r -> M=r / r+8
                if (m < NJ) out[tg * 98 + m * 2 + comp] = acc[r];
            }
        }
    }
}

extern "C" void kernel_launch(void* const* d_in, const int* in_sizes, int n_in,
                              void* d_out, int out_size, void* d_ws, size_t ws_size,
                              hipStream_t stream) {
    const float* thetas  = (const float*)d_in[0];  // local_thetas (T,49) f32
    const float* rootpos = (const float*)d_in[1];  // root_positions (T,49,2) f32
    const float* scales  = (const float*)d_in[2];  // bone_scales (T,49) f32
    const float* rest    = (const float*)d_in[3];  // rest_offsets (49,2) f32
    // d_in[4] = parents: arange-1 chain, structure hardcoded
    float* out = (float*)d_out;                    // (T,49,2) f32

    dim3 grid(T_TOTAL / TBLOCK);
    dim3 block(TBLOCK);
    fk2d_kernel<<<grid, block, 0, stream>>>(thetas, rootpos, scales, rest, out);
}